// _StackedGAT_55568286876148
// MI455X (gfx1250) — compile-verified
//
#include <hip/hip_runtime.h>
#include <math.h>

typedef __attribute__((ext_vector_type(2))) float v2f;
typedef __attribute__((ext_vector_type(8))) float v8f;

#define NEG_SLOPE 0.2f

// ============================================================================
// WMMA GEMM: out[N x 64] = A[N x K] @ W[K x 64], f32 via V_WMMA_F32_16X16X4_F32
// Block = 128 threads (4 wave32), 64 rows per block (4 row tiles).
// Wave w owns the 16-wide column tile [16w, 16w+16) and keeps its W slice in
// registers (K/4 x v2f); A staged in LDS with +4-float row padding so the
// half-wave column reads hit distinct banks. Steady state: ds_load + v_wmma.
// Operand layouts:
//   A (16x4 f32): lane l -> row = l&15, holds K = 4k + 2*(l>>4) + {0,1}
//   B (4x16 f32): lane l -> col = l&15, holds same K pair
//   C/D:          vgpr g -> M = g + (l>>4)*8, N = l&15
// ============================================================================
template <int K>
__global__ __launch_bounds__(128) void gemm_wmma_kernel(
    const float* __restrict__ A, const float* __restrict__ W,
    float* __restrict__ out, int N) {
  constexpr int KV = K / 4;    // k-steps (float4 vectors per row)
  constexpr int LDSK = K + 4;  // padded LDS row stride in floats (16B aligned)
  constexpr int ROWS = 64;     // rows per block
  __shared__ float sA[ROWS * LDSK];

  const int blockRow = blockIdx.x * ROWS;
  const int tid = threadIdx.x;

  // cooperative, coalesced staging of the (up to) 64 x K A tile into LDS
  for (int i = tid; i < ROWS * KV; i += 128) {
    const int r = i / KV, c = i % KV;
    float4 v = make_float4(0.f, 0.f, 0.f, 0.f);
    if (blockRow + r < N)
      v = ((const float4*)(A + (size_t)(blockRow + r) * K))[c];
    *(float4*)&sA[r * LDSK + c * 4] = v;
  }

  const int wave = tid >> 5;
  const int lane = tid & 31;
  const int hlf = lane >> 4;  // 0 or 1
  const int col = wave * 16 + (lane & 15);

  // register-resident 16-column slice of W for this wave
  v2f breg[KV];
#pragma unroll
  for (int k = 0; k < KV; ++k) {
    const int ka = k * 4 + hlf * 2;
    breg[k].x = W[(size_t)ka * 64 + col];
    breg[k].y = W[(size_t)(ka + 1) * 64 + col];
  }
  __syncthreads();

  for (int rt = 0; rt < ROWS / 16; ++rt) {
    const int tileRow = blockRow + rt * 16;
    if (tileRow >= N) break;  // N % 16 == 0: tile-granular guard, EXEC stays ~0
    const float* sRow = &sA[(rt * 16 + (lane & 15)) * LDSK];
    v8f acc = {};
#pragma unroll
    for (int k = 0; k < KV; ++k) {
      const int ka = k * 4 + hlf * 2;
      v2f a;
      a.x = sRow[ka];
      a.y = sRow[ka + 1];
      acc = __builtin_amdgcn_wmma_f32_16x16x4_f32(false, a, false, breg[k],
                                                  (short)0, acc, false, false);
    }
#pragma unroll
    for (int g = 0; g < 8; ++g)
      out[(size_t)(tileRow + g + hlf * 8) * 64 + col] = acc[g];
  }
}

// ============================================================================
// Per-node attention logits: ls[n] = h[n] . a_src, ld[n] = h[n] . a_dst
// ============================================================================
__global__ void logits_kernel(const float* __restrict__ h,
                              const float* __restrict__ asrc,
                              const float* __restrict__ adst,
                              float* __restrict__ ls, float* __restrict__ ld,
                              int N) {
  int n = blockIdx.x * blockDim.x + threadIdx.x;
  if (n >= N) return;
  const float4* hp = (const float4*)(h + (size_t)n * 64);
  const float4* ap = (const float4*)asrc;
  const float4* bp = (const float4*)adst;
  float s = 0.f, d = 0.f;
#pragma unroll
  for (int k = 0; k < 16; ++k) {
    float4 hv = hp[k], av = ap[k], bv = bp[k];
    s += hv.x * av.x + hv.y * av.y + hv.z * av.z + hv.w * av.w;
    d += hv.x * bv.x + hv.y * bv.y + hv.z * bv.z + hv.w * bv.w;
  }
  ls[n] = s;
  ld[n] = d;
}

// monotone float<->uint key mapping for atomic max of signed floats
__device__ __forceinline__ unsigned fkey(float f) {
  unsigned u = __float_as_uint(f);
  return (u & 0x80000000u) ? ~u : (u | 0x80000000u);
}
__device__ __forceinline__ float funkey(unsigned k) {
  return (k & 0x80000000u) ? __uint_as_float(k & 0x7FFFFFFFu)
                           : __uint_as_float(~k);
}

__device__ __forceinline__ void edge_sd(const long long* ei, long long e,
                                        int E, int& s, int& d) {
  if (e < (long long)E) {
    s = (int)ei[e];
    d = (int)ei[(long long)E + e];
  } else {  // appended self loops
    s = (int)(e - E);
    d = s;
  }
}

// pass 1: segment max over destinations (leaky-relu'd logits)
__global__ void edge_max_kernel(const long long* __restrict__ ei,
                                const float* __restrict__ ls,
                                const float* __restrict__ ld,
                                unsigned* __restrict__ mkey, int E, int N) {
  long long e = (long long)blockIdx.x * blockDim.x + threadIdx.x;
  if (e >= (long long)E + N) return;
  int s, d;
  edge_sd(ei, e, E, s, d);
  float x = ls[s] + ld[d];
  x = x > 0.f ? x : NEG_SLOPE * x;
  atomicMax(&mkey[d], fkey(x));
}

// pass 2: p = exp(e - max[dst]); denom[dst] += p
__global__ void edge_softmax_kernel(const long long* __restrict__ ei,
                                    const float* __restrict__ ls,
                                    const float* __restrict__ ld,
                                    const unsigned* __restrict__ mkey,
                                    float* __restrict__ pbuf,
                                    float* __restrict__ denom, int E, int N) {
  long long e = (long long)blockIdx.x * blockDim.x + threadIdx.x;
  if (e >= (long long)E + N) return;
  int s, d;
  edge_sd(ei, e, E, s, d);
  float x = ls[s] + ld[d];
  x = x > 0.f ? x : NEG_SLOPE * x;
  float p = __expf(x - funkey(mkey[d]));
  pbuf[e] = p;
  atomicAdd(&denom[d], p);
}

// pass 3: out[dst] += alpha * h[src]; one wave per edge, 2 features per lane.
__global__ void edge_aggregate_kernel(const long long* __restrict__ ei,
                                      const float* __restrict__ pbuf,
                                      const float* __restrict__ denom,
                                      const float* __restrict__ h,
                                      float* __restrict__ out, int E, int N) {
  long long t = (long long)blockIdx.x * blockDim.x + threadIdx.x;
  long long e = t >> 5;  // wave-uniform
  if (e >= (long long)E + N) return;
  int lane = threadIdx.x & 31;
  int s, d;
  edge_sd(ei, e, E, s, d);
  float alpha = pbuf[e] / (denom[d] + 1e-16f);
  float2 hv = ((const float2*)(h + (size_t)s * 64))[lane];
  atomicAdd(&out[(size_t)d * 64 + lane * 2 + 0], alpha * hv.x);
  atomicAdd(&out[(size_t)d * 64 + lane * 2 + 1], alpha * hv.y);
}

__global__ void bias_relu_kernel(float* __restrict__ h,
                                 const float* __restrict__ b, size_t total) {
  size_t i = (size_t)blockIdx.x * blockDim.x + threadIdx.x;
  if (i >= total) return;
  float v = h[i] + b[i & 63];
  h[i] = v > 0.f ? v : 0.f;
}

__global__ void classifier_kernel(const float* __restrict__ h,
                                  const float* __restrict__ Wc,
                                  const float* __restrict__ bc,
                                  float* __restrict__ y, int N) {
  int n = blockIdx.x * blockDim.x + threadIdx.x;
  if (n >= N) return;
  float a0 = bc[0], a1 = bc[1];
#pragma unroll 8
  for (int k = 0; k < 64; ++k) {
    float v = h[(size_t)n * 64 + k];
    a0 += v * Wc[k * 2 + 0];
    a1 += v * Wc[k * 2 + 1];
  }
  y[(size_t)n * 2 + 0] = a0;
  y[(size_t)n * 2 + 1] = a1;
}

extern "C" void kernel_launch(void* const* d_in, const int* in_sizes, int n_in,
                              void* d_out, int out_size, void* d_ws,
                              size_t ws_size, hipStream_t stream) {
  const float* x = (const float*)d_in[0];
  const long long* ei = (const long long*)d_in[1];  // int64 per reference
  const float* W0 = (const float*)d_in[2];
  const float* as0 = (const float*)d_in[3];
  const float* ad0 = (const float*)d_in[4];
  const float* b0 = (const float*)d_in[5];
  const float* Ws = (const float*)d_in[6];     // [2,64,64]
  const float* asrcs = (const float*)d_in[7];  // [2,64]
  const float* adsts = (const float*)d_in[8];  // [2,64]
  const float* bs = (const float*)d_in[9];     // [2,64]
  const float* Wc = (const float*)d_in[10];
  const float* bc = (const float*)d_in[11];

  const int N = in_sizes[0] / 128;  // 100000 (divisible by 16)
  const int E = in_sizes[1] / 2;    // 1600000
  const int Et = E + N;             // edges incl. self loops

  // workspace carve-up
  float* ws = (float*)d_ws;
  float* bufA = ws;  ws += (size_t)N * 64;  // linear projection h
  float* bufB = ws;  ws += (size_t)N * 64;  // aggregated output (ping)
  float* bufC = ws;  ws += (size_t)N * 64;  // aggregated output (pong)
  float* ls = ws;    ws += N;
  float* ld = ws;    ws += N;
  unsigned* mkey = (unsigned*)ws; ws += N;
  float* denom = ws; ws += N;
  float* pbuf = ws;  ws += Et;

  const int gemmBlocks = (N + 63) / 64;

  auto run_layer = [&](const float* in, int K, const float* W,
                       const float* asrc, const float* adst, const float* bias,
                       float* hl, float* out) {
    if (K == 128)
      gemm_wmma_kernel<128><<<gemmBlocks, 128, 0, stream>>>(in, W, hl, N);
    else
      gemm_wmma_kernel<64><<<gemmBlocks, 128, 0, stream>>>(in, W, hl, N);
    logits_kernel<<<(N + 255) / 256, 256, 0, stream>>>(hl, asrc, adst, ls, ld, N);
    hipMemsetAsync(mkey, 0, (size_t)N * 4, stream);  // 0 == minimal fkey
    hipMemsetAsync(denom, 0, (size_t)N * 4, stream);
    hipMemsetAsync(out, 0, (size_t)N * 64 * 4, stream);
    edge_max_kernel<<<(Et + 255) / 256, 256, 0, stream>>>(ei, ls, ld, mkey, E, N);
    edge_softmax_kernel<<<(Et + 255) / 256, 256, 0, stream>>>(ei, ls, ld, mkey,
                                                              pbuf, denom, E, N);
    long long aggT = (long long)Et * 32;
    edge_aggregate_kernel<<<(int)((aggT + 255) / 256), 256, 0, stream>>>(
        ei, pbuf, denom, hl, out, E, N);
    bias_relu_kernel<<<(int)(((size_t)N * 64 + 255) / 256), 256, 0, stream>>>(
        out, bias, (size_t)N * 64);
  };

  // layer 0: 128 -> 64
  run_layer(x, 128, W0, as0, ad0, b0, bufA, bufB);
  // layer 1: 64 -> 64
  run_layer(bufB, 64, Ws + 0 * 64 * 64, asrcs + 0 * 64, adsts + 0 * 64,
            bs + 0 * 64, bufA, bufC);
  // layer 2: 64 -> 64
  run_layer(bufC, 64, Ws + 1 * 64 * 64, asrcs + 1 * 64, adsts + 1 * 64,
            bs + 1 * 64, bufA, bufB);
  // classifier: 64 -> 2
  classifier_kernel<<<(N + 255) / 256, 256, 0, stream>>>(bufB, Wc, bc,
                                                         (float*)d_out, N);
}